// NPInstantaneousTransitionPrior_2087354106563
// MI455X (gfx1250) — compile-verified
//
#include <hip/hip_runtime.h>
#include <math.h>

// MI455X / gfx1250, wave32. FP32 pipeline -> V_WMMA_F32_16X16X4_F32.
// Compute-bound (~480 FLOP/byte): 20.5 GFLOP vs 43 MB of traffic.

typedef __attribute__((ext_vector_type(2))) float v2f;
typedef __attribute__((ext_vector_type(8))) float v8f;

#define SLOPE 0.2f
#define BPAD 68          // per-wave activation tile row pitch (dwords), even + bank-spread
#define W0PITCH 26       // W0 LDS row pitch (24 cols + pad)
#define WPITCH 66        // W1/W2 LDS row pitch (64 cols + pad)

__device__ __forceinline__ v8f wmma4(v2f a, v2f b, v8f c) {
    // V_WMMA_F32_16X16X4_F32 : D(16x16 f32) = A(16x4) x B(4x16) + C
    return __builtin_amdgcn_wmma_f32_16x16x4_f32(
        false, a, false, b, (short)0, c, false, false);
}

// A-frag: lane<16 -> row=lane, K={4kc,4kc+1}; lane>=16 -> row=lane-16, K={4kc+2,4kc+3}
__device__ __forceinline__ v2f afrag(const float* buf, int kc, int lane) {
    int row = lane & 15;
    int k   = (kc << 2) + ((lane >> 4) << 1);
    return *(const v2f*)(buf + row * BPAD + k);
}

// B-frag from transposed weight view: B[k][n] = W[n*stride + k]  (forward layers)
__device__ __forceinline__ v2f bfragT(const float* W, int stride, int kc, int t, int lane) {
    int n = (t << 4) + (lane & 15);
    int k = (kc << 2) + ((lane >> 4) << 1);
    return *(const v2f*)(W + n * stride + k);
}

// B-frag from natural weight view: B[k][n] = W[k*stride + n]  (backward layers)
__device__ __forceinline__ v2f bfragN(const float* W, int stride, int kc, int t, int lane) {
    int n = (t << 4) + (lane & 15);
    int k = (kc << 2) + ((lane >> 4) << 1);
    v2f r;
    r.x = W[k * stride + n];
    r.y = W[(k + 1) * stride + n];
    return r;
}

__global__ __launch_bounds__(128)
void prior_fused_kernel(const float* __restrict__ x,
                        const float* __restrict__ W0g, const float* __restrict__ b0g,
                        const float* __restrict__ W1g, const float* __restrict__ b1g,
                        const float* __restrict__ W2g, const float* __restrict__ b2g,
                        const float* __restrict__ Woutg, const float* __restrict__ boutg,
                        float* __restrict__ out, float* __restrict__ ws)
{
    __shared__ __align__(16) float W0s[64 * W0PITCH + 64]; // +64: natural-read n up to 31 pads past last row
    __shared__ __align__(16) float W1s[64 * WPITCH];
    __shared__ __align__(16) float W2s[64 * WPITCH];
    __shared__ __align__(16) float Wouts[64];
    __shared__ __align__(16) float b0s[64], b1s[64], b2s[64];
    __shared__ __align__(16) float bufA[4][16 * BPAD];
    __shared__ float sladS[4];

    const int tid  = threadIdx.x;
    const int lane = tid & 31;
    const int wv   = tid >> 5;
    const int z    = blockIdx.y;
    const int rblk = blockIdx.x * 64;       // first window row of this block
    const int b    = rblk >> 9;             // / 512 (T=512 divides 64-blocks evenly)
    const int t0   = rblk & 511;
    const int r0   = rblk + wv * 16;        // this wave's first row (global n)
    const int tw0  = t0 + wv * 16;          // this wave's first t within batch b

    // ---- stage weights for this z into LDS ----
    for (int i = tid; i < 64 * 24; i += 128) {
        int hh = i / 24, dd = i - hh * 24;
        W0s[hh * W0PITCH + dd] = W0g[(z * 64 + hh) * 24 + dd];
    }
    for (int i = tid; i < 64 * 64; i += 128) {
        int gg = i >> 6, hh = i & 63;
        W1s[gg * WPITCH + hh] = W1g[(z * 64 + gg) * 64 + hh];
        W2s[gg * WPITCH + hh] = W2g[(z * 64 + gg) * 64 + hh];
    }
    if (tid < 64) {
        Wouts[tid] = Woutg[z * 64 + tid];
        b0s[tid]   = b0g[z * 64 + tid];
        b1s[tid]   = b1g[z * 64 + tid];
        b2s[tid]   = b2g[z * 64 + tid];
    }
    __syncthreads();

    float* myBuf = bufA[wv];

    // ---- load 16 windows (16 x 24 contiguous floats each) into activation tile ----
    for (int i = 0; i < 12; ++i) {
        int idx = i * 32 + lane;            // 384 = 12*32 elements
        int row = idx / 24, col = idx - row * 24;
        myBuf[row * BPAD + col] = x[((size_t)(b * 514 + tw0 + row)) * 8 + col];
    }

    // ---- layer 0: h0(16x64) = full(16x24) x W0^T, leaky ----
    v2f a6[6];
    #pragma unroll
    for (int kc = 0; kc < 6; ++kc) a6[kc] = afrag(myBuf, kc, lane);
    float d0r[4][8];
    #pragma unroll
    for (int t = 0; t < 4; ++t) {
        v8f c = {};
        #pragma unroll
        for (int kc = 0; kc < 6; ++kc)
            c = wmma4(a6[kc], bfragT(W0s, W0PITCH, kc, t, lane), c);
        float bias = b0s[(t << 4) + (lane & 15)];
        #pragma unroll
        for (int v = 0; v < 8; ++v) {
            float hv = c[v] + bias;
            float dv = hv > 0.0f ? 1.0f : SLOPE;
            d0r[t][v] = dv;
            myBuf[(v + ((lane >> 4) << 3)) * BPAD + (t << 4) + (lane & 15)] = hv * dv;
        }
    }

    // ---- layer 1 ----
    v2f a16[16];
    #pragma unroll
    for (int kc = 0; kc < 16; ++kc) a16[kc] = afrag(myBuf, kc, lane);
    float d1r[4][8];
    #pragma unroll
    for (int t = 0; t < 4; ++t) {
        v8f c = {};
        #pragma unroll
        for (int kc = 0; kc < 16; ++kc)
            c = wmma4(a16[kc], bfragT(W1s, WPITCH, kc, t, lane), c);
        float bias = b1s[(t << 4) + (lane & 15)];
        #pragma unroll
        for (int v = 0; v < 8; ++v) {
            float hv = c[v] + bias;
            float dv = hv > 0.0f ? 1.0f : SLOPE;
            d1r[t][v] = dv;
            myBuf[(v + ((lane >> 4) << 3)) * BPAD + (t << 4) + (lane & 15)] = hv * dv;
        }
    }

    // ---- layer 2 + out-dot + g2 = d2*Wout ----
    #pragma unroll
    for (int kc = 0; kc < 16; ++kc) a16[kc] = afrag(myBuf, kc, lane);
    float p[8];
    #pragma unroll
    for (int v = 0; v < 8; ++v) p[v] = 0.0f;
    #pragma unroll
    for (int t = 0; t < 4; ++t) {
        v8f c = {};
        #pragma unroll
        for (int kc = 0; kc < 16; ++kc)
            c = wmma4(a16[kc], bfragT(W2s, WPITCH, kc, t, lane), c);
        float bias = b2s[(t << 4) + (lane & 15)];
        float wo   = Wouts[(t << 4) + (lane & 15)];
        #pragma unroll
        for (int v = 0; v < 8; ++v) {
            float hv = c[v] + bias;
            float dv = hv > 0.0f ? 1.0f : SLOPE;
            p[v] += hv * dv * wo;                                   // a2 * Wout
            myBuf[(v + ((lane >> 4) << 3)) * BPAD + (t << 4) + (lane & 15)] = dv * wo; // g2
        }
    }
    // reduce out across the 16 column-lanes of each half-wave
    #pragma unroll
    for (int v = 0; v < 8; ++v) {
        float s = p[v];
        s += __shfl_xor(s, 1);
        s += __shfl_xor(s, 2);
        s += __shfl_xor(s, 4);
        s += __shfl_xor(s, 8);
        p[v] = s;
    }
    if ((lane & 15) == 0) {
        float bo = boutg[z];
        #pragma unroll
        for (int v = 0; v < 8; ++v) {
            int row = v + ((lane >> 4) << 3);
            out[(size_t)(r0 + row) * 8 + z] = p[v] + bo;            // residuals[b,t,z]
        }
    }

    // ---- backward: g1 = (g2 @ W2) * d1 ----
    #pragma unroll
    for (int kc = 0; kc < 16; ++kc) a16[kc] = afrag(myBuf, kc, lane);
    #pragma unroll
    for (int t = 0; t < 4; ++t) {
        v8f c = {};
        #pragma unroll
        for (int kc = 0; kc < 16; ++kc)
            c = wmma4(a16[kc], bfragN(W2s, WPITCH, kc, t, lane), c);
        #pragma unroll
        for (int v = 0; v < 8; ++v)
            myBuf[(v + ((lane >> 4) << 3)) * BPAD + (t << 4) + (lane & 15)] = c[v] * d1r[t][v];
    }

    // ---- g0 = (g1 @ W1) * d0 ----
    #pragma unroll
    for (int kc = 0; kc < 16; ++kc) a16[kc] = afrag(myBuf, kc, lane);
    #pragma unroll
    for (int t = 0; t < 4; ++t) {
        v8f c = {};
        #pragma unroll
        for (int kc = 0; kc < 16; ++kc)
            c = wmma4(a16[kc], bfragN(W1s, WPITCH, kc, t, lane), c);
        #pragma unroll
        for (int v = 0; v < 8; ++v)
            myBuf[(v + ((lane >> 4) << 3)) * BPAD + (t << 4) + (lane & 15)] = c[v] * d0r[t][v];
    }

    // ---- jac(16x24) = g0 @ W0 ; hist_jac stores + log|diag| partial ----
    #pragma unroll
    for (int kc = 0; kc < 16; ++kc) a16[kc] = afrag(myBuf, kc, lane);
    const int zc = 16 + z;
    const size_t off = (size_t)524416 + (size_t)65536 * (size_t)(16 * z + (z * (z - 1)) / 2);
    float lsum = 0.0f;
    #pragma unroll
    for (int t = 0; t < 2; ++t) {
        v8f c = {};
        #pragma unroll
        for (int kc = 0; kc < 16; ++kc)
            c = wmma4(a16[kc], bfragN(W0s, W0PITCH, kc, t, lane), c);
        int col = (t << 4) + (lane & 15);
        if (col < zc) {
            #pragma unroll
            for (int v = 0; v < 8; ++v) {
                int row = v + ((lane >> 4) << 3);
                out[off + (size_t)(r0 + row) * zc + col] = c[v];
            }
        }
        if (t == 1 && (lane & 15) == z) {       // diag column 16+z lives in tile 1
            #pragma unroll
            for (int v = 0; v < 8; ++v) lsum += logf(fabsf(c[v]));
        }
    }
    lsum += __shfl_xor(lsum, 16);               // combine rows 0-7 with rows 8-15
    if (lane == z) sladS[wv] = lsum;
    __syncthreads();
    if (tid == 0)
        ws[(size_t)z * 1024 + blockIdx.x] = sladS[0] + sladS[1] + sladS[2] + sladS[3];
}

__global__ __launch_bounds__(64)
void reduce_slad_kernel(const float* __restrict__ ws, float* __restrict__ out)
{
    __shared__ float s[64];
    const int b = blockIdx.x, tid = threadIdx.x;
    const int z = tid >> 3, j = tid & 7;
    s[tid] = ws[(size_t)z * 1024 + b * 8 + j];
    __syncthreads();
    for (int off = 32; off > 0; off >>= 1) {
        if (tid < off) s[tid] += s[tid + off];
        __syncthreads();
    }
    if (tid == 0) out[524288 + b] = s[0];
}

extern "C" void kernel_launch(void* const* d_in, const int* in_sizes, int n_in,
                              void* d_out, int out_size, void* d_ws, size_t ws_size,
                              hipStream_t stream) {
    const float* x    = (const float*)d_in[0];
    const float* W0   = (const float*)d_in[1];
    const float* b0   = (const float*)d_in[2];
    const float* W1   = (const float*)d_in[3];
    const float* b1   = (const float*)d_in[4];
    const float* W2   = (const float*)d_in[5];
    const float* b2   = (const float*)d_in[6];
    const float* Wout = (const float*)d_in[7];
    const float* bout = (const float*)d_in[8];
    float* out = (float*)d_out;
    float* ws  = (float*)d_ws;   // 8192 floats of per-block log|det| partials

    dim3 grid(1024, 8);          // 1024 row-blocks (64 windows each) x 8 z-slices
    prior_fused_kernel<<<grid, 128, 0, stream>>>(x, W0, b0, W1, b1, W2, b2, Wout, bout, out, ws);
    reduce_slad_kernel<<<128, 64, 0, stream>>>(ws, out);
}